// Izhikevich_27513560498764
// MI455X (gfx1250) — compile-verified
//
#include <hip/hip_runtime.h>
#include <stdint.h>

// Problem constants from the reference: x is (B, T, N) float32, B=64, T=256, N=2048.
#define B_DIM 64
#define T_STEPS 256
#define N_NEUR 2048
#define BLOCK 256          // threads per block (8 wave32s); 256 divides N_NEUR
#define DEPTH 16           // async-to-LDS pipeline depth (must be power of two)
#define PF_DIST 64         // global_prefetch lookahead (time steps) for HBM -> L2

__global__ __launch_bounds__(BLOCK) void izhikevich_async_kernel(
    const float* __restrict__ x, const float* __restrict__ a_p,
    const float* __restrict__ b_p, const float* __restrict__ c_p,
    const float* __restrict__ d_p, float* __restrict__ out) {
  // LDS ring buffer: DEPTH slots x BLOCK lanes x 4B = 16 KB per workgroup.
  __shared__ float buf[DEPTH * BLOCK];

  const int tid = threadIdx.x;
  const int blocksPerB = N_NEUR / BLOCK;  // 8, so bi/n0 are block-uniform
  const int bi = blockIdx.x / blocksPerB;
  const int n0 = (blockIdx.x % blocksPerB) * BLOCK;

  // Uniform scalars (SMEM loads).
  const float a_s = a_p[0];
  const float b_s = b_p[0];
  const float c_v = c_p[0];
  const float d_v = d_p[0];

  const size_t base_elt = (size_t)bi * T_STEPS * N_NEUR + (size_t)n0;
  const float* xbase = x + base_elt;          // block-uniform -> SGPR pair
  const unsigned long long xb64 = (unsigned long long)(uintptr_t)xbase;
  float* op = out + base_elt + tid;           // this lane's output stream

  const unsigned row_bytes = (unsigned)(N_NEUR * 4);     // 8192 B per time step
  const unsigned lane_off = (unsigned)(tid * 4);         // lane's byte offset in a row
  // Wave-relative LDS byte address of this lane's slot-0 element
  // (low 32 bits of the generic pointer are the LDS offset on gfx1250).
  const unsigned lds0 = (unsigned)(uintptr_t)&buf[tid];
  const unsigned slot_bytes = (unsigned)(BLOCK * 4);     // 1024 B per slot

  // ---- Prologue: fill all DEPTH pipeline slots with t = 0..DEPTH-1 ----
#pragma unroll
  for (int j = 0; j < DEPTH; ++j) {
    unsigned ldsa = lds0 + (unsigned)j * slot_bytes;
    unsigned goff = lane_off + (unsigned)j * row_bytes;
    asm volatile("global_load_async_to_lds_b32 %0, %1, %2"
                 :: "v"(ldsa), "v"(goff), "s"(xb64)
                 : "memory");
  }

  // Izhikevich state.
  float v = -65.0f;
  float u = -65.0f * b_s;

  int slot = 0;
  unsigned voff_issue = lane_off + (unsigned)DEPTH * row_bytes;  // byte off of t+DEPTH

  // ---- Steady state: wait-oldest / consume / refill ----
#pragma unroll 4
  for (int t = 0; t < T_STEPS - DEPTH; ++t) {
    // Async loads complete in order: ASYNCcnt <= DEPTH-1 => load for step t landed.
    asm volatile("s_wait_asynccnt %0" :: "i"(DEPTH - 1) : "memory");
    float xt = buf[slot * BLOCK + tid];
    // Ensure the LDS read retired before the DMA engine may overwrite this slot (WAR).
    asm volatile("s_wait_dscnt 0" ::: "memory");

    // Refill this slot with step t+DEPTH.
    unsigned ldsa = lds0 + (unsigned)slot * slot_bytes;
    asm volatile("global_load_async_to_lds_b32 %0, %1, %2"
                 :: "v"(ldsa), "v"(voff_issue), "s"(xb64)
                 : "memory");
    voff_issue += row_bytes;

    // Deep HBM -> L2 prefetch (no counters, no registers held).
    if (t + PF_DIST < T_STEPS)
      __builtin_prefetch(xbase + (size_t)(t + PF_DIST) * N_NEUR + tid, 0, 3);

    // Izhikevich update (matches reference ordering; branchless reset).
    v = v + 0.04f * v * v + 5.0f * v + 140.0f - u + xt;
    u = u + a_s * (b_s * v - u);
    const bool fired = v >= 30.0f;
    *op = fired ? 1.0f : 0.0f;
    v = fired ? c_v : v;
    u = fired ? (u + d_v) : u;
    op += N_NEUR;

    slot = (slot + 1) & (DEPTH - 1);
  }

  // ---- Tail: drain the remaining DEPTH slots ----
  asm volatile("s_wait_asynccnt 0" ::: "memory");
#pragma unroll
  for (int t = T_STEPS - DEPTH; t < T_STEPS; ++t) {
    float xt = buf[slot * BLOCK + tid];
    v = v + 0.04f * v * v + 5.0f * v + 140.0f - u + xt;
    u = u + a_s * (b_s * v - u);
    const bool fired = v >= 30.0f;
    *op = fired ? 1.0f : 0.0f;
    v = fired ? c_v : v;
    u = fired ? (u + d_v) : u;
    op += N_NEUR;
    slot = (slot + 1) & (DEPTH - 1);
  }
}

extern "C" void kernel_launch(void* const* d_in, const int* in_sizes, int n_in,
                              void* d_out, int out_size, void* d_ws, size_t ws_size,
                              hipStream_t stream) {
  (void)in_sizes; (void)n_in; (void)d_ws; (void)ws_size; (void)out_size;
  const float* x = (const float*)d_in[0];
  const float* a = (const float*)d_in[1];
  const float* b = (const float*)d_in[2];
  const float* c = (const float*)d_in[3];
  const float* dd = (const float*)d_in[4];
  float* out = (float*)d_out;

  dim3 grid((B_DIM * N_NEUR) / BLOCK);  // 512 blocks = 4096 wave32s
  dim3 block(BLOCK);
  hipLaunchKernelGGL(izhikevich_async_kernel, grid, block, 0, stream,
                     x, a, b, c, dd, out);
}